// Encoder_35424890257737
// MI455X (gfx1250) — compile-verified
//
#include <hip/hip_runtime.h>

typedef __attribute__((ext_vector_type(2))) float v2f;
typedef __attribute__((ext_vector_type(8))) float v8f;

// ---------------------------------------------------------------------------
// init: deg[0..N) = 1.0 (self-loop weight), accumulators A1/A2 = 0
// ws float layout: [deg N][A1 N*128][A2 N*64][H1 N*128][H2 N*64]
// ---------------------------------------------------------------------------
__global__ __launch_bounds__(256) void init_ws_kernel(float* __restrict__ ws, int n, long total) {
    long i = (long)blockIdx.x * blockDim.x + threadIdx.x;
    if (i < total) ws[i] = (i < n) ? 1.0f : 0.0f;
}

__global__ __launch_bounds__(256) void degree_kernel(const int* __restrict__ dst,
                                                     float* __restrict__ deg, int nE) {
    int i = blockIdx.x * blockDim.x + threadIdx.x;
    if (i < nE) atomicAdd(&deg[dst[i]], 1.0f);
}

__global__ __launch_bounds__(256) void rsqrt_kernel(float* __restrict__ deg, int n) {
    int i = blockIdx.x * blockDim.x + threadIdx.x;
    if (i < n) deg[i] = 1.0f / sqrtf(deg[i]);
}

// ---------------------------------------------------------------------------
// H[nRows x OUTCOLS] = X[nRows x K] @ W[K x OUTCOLS], exact fp32 via
// V_WMMA_F32_16X16X4_F32. One wave -> one 16x16 output tile, K-loop step 4,
// fully unrolled with compile-time strides so every load is base+imm-offset.
//   A frag: lane l holds row m=l&15, k-pair base kg=2*(l>>4) -> float2 loads.
//   B frag: lane l holds col n=l&15, same k-pair; lane-dependent kg folded
//           into the base pointer once, loop offsets are immediates.
//   D frag: VGPR v, lane l -> row v + 8*(l>>4), col l&15.
// ---------------------------------------------------------------------------
template <int K, int OUTCOLS>
__global__ __launch_bounds__(256) void gemm_wmma_kernel(const float* __restrict__ X,
                                                        const float* __restrict__ W,
                                                        float* __restrict__ H,
                                                        int totalTiles) {
    constexpr int NCT = OUTCOLS / 16;
    int wave = threadIdx.x >> 5;
    int lane = threadIdx.x & 31;
    int tile = blockIdx.x * 8 + wave;
    if (tile >= totalTiles) return;          // whole-wave exit: EXEC stays all-1s for WMMA
    int rowTile = tile / NCT;
    int colTile = tile - rowTile * NCT;

    int m  = lane & 15;
    int kg = (lane >> 4) << 1;               // 0 or 2

    const float* xrow = X + (size_t)(rowTile * 16 + m) * K + kg;
    const float* wcol = W + (size_t)kg * OUTCOLS + colTile * 16 + m;

    v8f acc = {};
#pragma unroll
    for (int k0 = 0; k0 < K; k0 += 4) {
        v2f a = *(const v2f*)(xrow + k0);            // 8B aligned (kg even, K mult of 4)
        v2f b;
        b.x = wcol[k0 * OUTCOLS];                    // imm offset: k0*OUTCOLS*4 bytes
        b.y = wcol[k0 * OUTCOLS + OUTCOLS];
        acc = __builtin_amdgcn_wmma_f32_16x16x4_f32(false, a, false, b,
                                                    (short)0, acc, false, false);
    }

    float* hbase = H + (size_t)(rowTile * 16) * OUTCOLS + colTile * 16 + m
                     + (size_t)((lane >> 4) << 3) * OUTCOLS;
#pragma unroll
    for (int v = 0; v < 8; ++v)
        hbase[v * OUTCOLS] = acc[v];
}

// ---------------------------------------------------------------------------
// One wave per edge: coalesced float4/float2 gather of H[src], per-edge norm,
// fp32 atomic scatter-add into Acc[dst] (resolved at L2).
// ---------------------------------------------------------------------------
template <int F>
__global__ __launch_bounds__(256) void scatter_kernel(const float* __restrict__ H,
                                                      const int* __restrict__ src,
                                                      const int* __restrict__ dst,
                                                      const float* __restrict__ dinv,
                                                      float* __restrict__ Acc, int nE) {
    int e    = blockIdx.x * 8 + (threadIdx.x >> 5);
    int lane = threadIdx.x & 31;
    if (e >= nE) return;
    int s = src[e], d = dst[e];
    float norm = dinv[s] * dinv[d];
    constexpr int V = F / 32;
    const float* hs = H   + (size_t)s * F + lane * V;
    float*       ad = Acc + (size_t)d * F + lane * V;
    if constexpr (V == 4) {
        float4 h = *(const float4*)hs;
        atomicAdd(ad + 0, h.x * norm);
        atomicAdd(ad + 1, h.y * norm);
        atomicAdd(ad + 2, h.z * norm);
        atomicAdd(ad + 3, h.w * norm);
    } else {
        float2 h = *(const float2*)hs;
        atomicAdd(ad + 0, h.x * norm);
        atomicAdd(ad + 1, h.y * norm);
    }
}

// ---------------------------------------------------------------------------
// Out = relu(Acc + H*dinv^2 + bias)   (self-loop + bias + activation, fused)
// ---------------------------------------------------------------------------
template <int F>
__global__ __launch_bounds__(256) void combine_kernel(const float* __restrict__ Acc,
                                                      const float* __restrict__ Hin,
                                                      const float* __restrict__ dinv,
                                                      const float* __restrict__ bias,
                                                      float* __restrict__ Out, int n) {
    int i = blockIdx.x * blockDim.x + threadIdx.x;
    if (i >= n * F) return;
    int node = i / F;              // F is a power of two -> shift
    int f    = i & (F - 1);
    float di = dinv[node];
    float v  = Acc[i] + Hin[i] * (di * di) + bias[f];
    Out[i] = v > 0.0f ? v : 0.0f;
}

extern "C" void kernel_launch(void* const* d_in, const int* in_sizes, int n_in,
                              void* d_out, int out_size, void* d_ws, size_t ws_size,
                              hipStream_t stream) {
    const float* x  = (const float*)d_in[0];
    const int*   ei = (const int*)d_in[1];
    const float* W1 = (const float*)d_in[2];
    const float* b1 = (const float*)d_in[3];
    const float* W2 = (const float*)d_in[4];
    const float* b2 = (const float*)d_in[5];
    float* out = (float*)d_out;

    const int DIN = 128, DHID = 128, DOUT = 64;
    const int N = in_sizes[0] / DIN;     // 50000 (divisible by 16)
    const int E = in_sizes[1] / 2;       // 800000
    const int* src = ei;                 // edge_index[0]
    const int* dst = ei + E;             // edge_index[1]

    float* ws  = (float*)d_ws;
    float* deg = ws;                                 // N  (becomes dinv after rsqrt)
    float* A1  = deg + N;                            // N*DHID
    float* A2  = A1 + (size_t)N * DHID;              // N*DOUT
    float* H1  = A2 + (size_t)N * DOUT;              // N*DHID
    float* H2  = H1 + (size_t)N * DHID;              // N*DOUT

    // init deg=1 (self loop) and zero accumulators (contiguous region)
    long initTotal = (long)N * (1 + DHID + DOUT);
    init_ws_kernel<<<(int)((initTotal + 255) / 256), 256, 0, stream>>>(ws, N, initTotal);
    degree_kernel<<<(E + 255) / 256, 256, 0, stream>>>(dst, deg, E);
    rsqrt_kernel<<<(N + 255) / 256, 256, 0, stream>>>(deg, N);

    int edgeBlocks = (E + 7) / 8;        // one wave per edge, 8 waves/block

    // ---- layer 1: H1 = X @ W1 ; A1 = scatter ; H1 = relu(A1 + H1*dinv^2 + b1)
    int tiles1 = (N / 16) * (DHID / 16);
    gemm_wmma_kernel<128, 128><<<(tiles1 + 7) / 8, 256, 0, stream>>>(x, W1, H1, tiles1);
    scatter_kernel<128><<<edgeBlocks, 256, 0, stream>>>(H1, src, dst, deg, A1, E);
    combine_kernel<128><<<(N * DHID + 255) / 256, 256, 0, stream>>>(A1, H1, deg, b1, H1, N);

    // ---- layer 2: H2 = H1 @ W2 ; A2 = scatter ; out = relu(A2 + H2*dinv^2 + b2)
    int tiles2 = (N / 16) * (DOUT / 16);
    gemm_wmma_kernel<128, 64><<<(tiles2 + 7) / 8, 256, 0, stream>>>(H1, W2, H2, tiles2);
    scatter_kernel<64><<<edgeBlocks, 256, 0, stream>>>(H2, src, dst, deg, A2, E);
    combine_kernel<64><<<(N * DOUT + 255) / 256, 256, 0, stream>>>(A2, H2, deg, b2, out, N);
}